// Gnn_layer_44495861186887
// MI455X (gfx1250) — compile-verified
//
#include <hip/hip_runtime.h>

// CDNA5 / gfx1250, wave32.
typedef __attribute__((ext_vector_type(2))) float v2f;
typedef __attribute__((ext_vector_type(8))) float v8f;

#define XS_STRIDE 132    // 128 + 4 pad: A-frag ds_load_b64 conflict-free
#define WTP_STRIDE 288   // pair-row stride (words), == 32 mod 64 -> conflict-free

// ---------------------------------------------------------------------------
// Pass A: message reductions
// msg_m[b,m,d] = sum_k X[b,m,k,d]   (contiguous 64x128 block per (b,m))
__global__ __launch_bounds__(128) void reduce_m_kernel(
    const float* __restrict__ X, float* __restrict__ msg_m)
{
  const int bm = blockIdx.x;                 // b*64 + m, 4096 blocks
  const int t  = threadIdx.x;                // d
  const float* base = X + (size_t)bm * 64 * 128;
  float s = 0.f;
#pragma unroll 8
  for (int k = 0; k < 64; ++k) s += base[k * 128 + t];
  msg_m[(size_t)bm * 128 + t] = s;
}

// msg_k[b,k,d] = sum_m X[b,m,k,d]
__global__ __launch_bounds__(128) void reduce_k_kernel(
    const float* __restrict__ X, float* __restrict__ msg_k)
{
  const int bk = blockIdx.x;                 // b*64 + k, 4096 blocks
  const int b  = bk >> 6;
  const int k  = bk & 63;
  const int t  = threadIdx.x;                // d
  const float* base = X + ((size_t)b * 64 * 64 + k) * 128;
  float s = 0.f;
#pragma unroll 8
  for (int m = 0; m < 64; ++m) s += base[(size_t)m * 64 * 128 + t];
  msg_k[(size_t)bk * 128 + t] = s;
}

// ---------------------------------------------------------------------------
// WMMA f32 GEMM: out[r, n] = sum_d X[r, d] * W[n, d]  (+ optional broadcast
// bias zm[bm, n] + zk[bk, n] decoded from r when fuse != 0; fused output uses
// non-temporal stores so the 134MB result streams to HBM without evicting the
// L2-resident input tensor).
// Block = 256 threads = 8 waves; block covers 128 rows, each wave 16 rows x
// full N=128 (8 WMMA tiles of 16x16, K-loop of 32 steps of 4).
__global__ __launch_bounds__(256) void gemm128_wmma_kernel(
    const float* __restrict__ X, const float* __restrict__ W,
    float* __restrict__ out,
    const float* __restrict__ zm, const float* __restrict__ zk,
    int fuse)
{
  extern __shared__ float lds[];
  float* Xs  = lds;                        // [128][XS_STRIDE]
  float* WTp = lds + 128 * XS_STRIDE;      // [64][WTP_STRIDE]:
                                           //   WTp[kp*288 + 2n + j] = W[n][2kp+j]

  const int t = threadIdx.x;
  const size_t row_base = (size_t)blockIdx.x * 128;

  // Stage 128x128 X tile via float4 (coalesced global, b128 LDS stores).
  {
    const float4* Xg = (const float4*)(X + row_base * 128);
#pragma unroll
    for (int j = 0; j < 16; ++j) {
      int idx = t + j * 256;               // float4 index, 4096 total
      int row = idx >> 5;                  // 32 float4 per row
      int c4  = idx & 31;
      float4 v = Xg[idx];
      *(float4*)(Xs + row * XS_STRIDE + c4 * 4) = v;
    }
  }
  // Stage W in pair-interleaved transposed layout. Consecutive lanes write
  // consecutive n at fixed kp -> conflict-free b64 LDS stores. Global side is
  // a small gathered float2 read of the 64KB weight matrix (L2-resident).
  {
    const float2* W2 = (const float2*)W;   // W2[n*64 + kp] = (W[n][2kp], W[n][2kp+1])
#pragma unroll
    for (int j = 0; j < 32; ++j) {
      int idx = t + j * 256;               // 8192 float2 total
      int n  = idx & 127;
      int kp = idx >> 7;                   // 0..63
      float2 w = W2[n * 64 + kp];
      *(float2*)(WTp + kp * WTP_STRIDE + 2 * n) = w;
    }
  }
  __syncthreads();

  const int wave = t >> 5;
  const int lane = t & 31;
  const int lm   = lane & 15;              // 0..15
  const int lg   = lane >> 4;              // 0..1

  v8f acc[8] = {};                         // 8 N-tiles of 16x16 f32 C

  // A fragment (ISA 7.12.2, 32-bit A 16x4): lane L, vgpr v ->
  //   A[m = L%16][k = v + 2*(L/16)]
  const float* arow = Xs + (wave * 16 + lm) * XS_STRIDE + 2 * lg;
  // B fragment (4x16): lane L, vgpr v -> B[k = v + 2*(L/16)][n = L%16].
  // Pair (k, k+1) for k = 4kt + 2lg lives at pair-row kp = 2kt + lg, cols
  // (2n, 2n+1) -> single aligned v2f (b64) load per fragment, no repacking.
  const float* brow = WTp + lg * WTP_STRIDE + 2 * lm;

  for (int kt = 0; kt < 32; ++kt) {
    v2f a = *(const v2f*)(arow + 4 * kt);  // 8B-aligned
    const float* bk = brow + kt * (2 * WTP_STRIDE);
#pragma unroll
    for (int nt = 0; nt < 8; ++nt) {
      v2f b = *(const v2f*)(bk + nt * 32);
      acc[nt] = __builtin_amdgcn_wmma_f32_16x16x4_f32(
          /*neg_a=*/false, a, /*neg_b=*/false, b,
          /*c_mod=*/(short)0, acc[nt],
          /*reuse_a=*/false, /*reuse_b=*/false);
    }
  }

  // Epilogue. C layout (16x16 f32): vgpr i, lane L -> row = i + 8*(L/16),
  // col = L%16. Row r decodes as r = ((b*64+m)*64 + k).
  const size_t r0 = row_base + wave * 16 + 8 * lg;
#pragma unroll
  for (int nt = 0; nt < 8; ++nt) {
    const int col = nt * 16 + lm;
#pragma unroll
    for (int i = 0; i < 8; ++i) {
      const size_t r = r0 + i;
      float val = acc[nt][i];
      if (fuse) {
        const size_t bm = r >> 6;                          // b*64 + m
        const size_t bk2 = ((r >> 12) << 6) | (r & 63);    // b*64 + k
        val += zm[bm * 128 + col] + zk[bk2 * 128 + col];
        __builtin_nontemporal_store(val, &out[r * 128 + col]);
      } else {
        out[r * 128 + col] = val;
      }
    }
  }
}

// ---------------------------------------------------------------------------
extern "C" void kernel_launch(void* const* d_in, const int* in_sizes, int n_in,
                              void* d_out, int out_size, void* d_ws, size_t ws_size,
                              hipStream_t stream)
{
  (void)in_sizes; (void)n_in; (void)out_size; (void)ws_size;
  const float* X     = (const float*)d_in[0];   // (64,64,64,128)
  const float* Wself = (const float*)d_in[1];   // (128,128)
  const float* Wm    = (const float*)d_in[2];
  const float* Wk    = (const float*)d_in[3];
  float* out = (float*)d_out;

  const size_t MSG = (size_t)4096 * 128;        // 0.5M floats = 2MB each
  float* msg_m = (float*)d_ws;
  float* msg_k = msg_m + MSG;
  float* zm    = msg_k + MSG;
  float* zk    = zm + MSG;

  const size_t lds_bytes =
      (size_t)(128 * XS_STRIDE + 64 * WTP_STRIDE) * sizeof(float);

  // Pass A: reductions over k (msg_m) and m (msg_k). Second pass mostly hits
  // the 192MB L2 with the input left resident by the first.
  reduce_m_kernel<<<4096, 128, 0, stream>>>(X, msg_m);
  reduce_k_kernel<<<4096, 128, 0, stream>>>(X, msg_k);

  // Pass B: small GEMMs for the broadcast message terms (4096 rows -> 32 blocks).
  gemm128_wmma_kernel<<<32, 256, lds_bytes, stream>>>(msg_m, Wm, zm, nullptr, nullptr, 0);
  gemm128_wmma_kernel<<<32, 256, lds_bytes, stream>>>(msg_k, Wk, zk, nullptr, nullptr, 0);

  // Pass C: main fused GEMM (262144 rows -> 2048 blocks), NT output stores.
  gemm128_wmma_kernel<<<2048, 256, lds_bytes, stream>>>(X, Wself, out, zm, zk, 1);
}